// ResGCN_12824772345977
// MI455X (gfx1250) — compile-verified
//
#include <hip/hip_runtime.h>
#include <math.h>

typedef __attribute__((ext_vector_type(2))) float v2f;
typedef __attribute__((ext_vector_type(8))) float v8f;

#define HID 128
#define NNODES 50000
#define NEDGES 800000
#define NGRAPHS 512
#define MAXHOP 8
#define BN_EPS 1e-5f

// ---------------------------------------------------------------------------
// fp32 WMMA 16x16x4 wrapper (D = A(16x4) x B(4x16) + C)
// ---------------------------------------------------------------------------
__device__ __forceinline__ v8f wmma4f32(v2f a, v2f b, v8f c) {
  return __builtin_amdgcn_wmma_f32_16x16x4_f32(
      /*neg_a=*/false, a, /*neg_b=*/false, b,
      /*c_mod=*/(short)0, c, /*reuse_a=*/false, /*reuse_b=*/false);
}

__device__ __forceinline__ void atomAddF(float* p, float v) {
  unsafeAtomicAdd(p, v);
}

// ---------------------------------------------------------------------------
// Generic helpers
// ---------------------------------------------------------------------------
__global__ void zero_kernel(float* __restrict__ p, int n) {
  int i = blockIdx.x * blockDim.x + threadIdx.x;
  if (i < n) p[i] = 0.f;
}

__global__ void fill_kernel(float* __restrict__ p, float v, int n) {
  int i = blockIdx.x * blockDim.x + threadIdx.x;
  if (i < n) p[i] = v;
}

// column sums + sums of squares of A[M,128] into sums[0:128]=sum, sums[128:256]=sumsq
__global__ void colreduce_kernel(const float* __restrict__ A, int M,
                                 float* __restrict__ sums) {
  int col = threadIdx.x;                 // blockDim.x == 128
  int r0 = blockIdx.x * 256;
  float s = 0.f, s2 = 0.f;
  for (int r = 0; r < 256; ++r) {
    int row = r0 + r;
    if (row < M) {
      float v = A[(size_t)row * HID + col];
      s += v; s2 += v * v;
    }
  }
  atomAddF(&sums[col], s);
  atomAddF(&sums[HID + col], s2);
}

// Fold BN(mean/var from sums, gamma, beta) into W[128,128] -> Weff, beff(+extra_b)
__global__ void fold_kernel(const float* __restrict__ sums,
                            const float* __restrict__ gamma,
                            const float* __restrict__ bbeta,
                            const float* __restrict__ W,
                            const float* __restrict__ extra_b, int M,
                            float* __restrict__ Weff, float* __restrict__ beff) {
  int n = threadIdx.x;                   // blockDim.x == 128, one block
  float invM = 1.f / (float)M;
  float acc = extra_b ? extra_b[n] : 0.f;
  for (int k = 0; k < HID; ++k) {
    float mean = sums[k] * invM;
    float var  = sums[HID + k] * invM - mean * mean;
    float sc = gamma[k] * rsqrtf(var + BN_EPS);
    float sh = bbeta[k] - mean * sc;
    float wkn = W[k * HID + n];
    Weff[k * HID + n] = sc * wkn;
    acc += sh * wkn;
  }
  beff[n] = acc;
}

// ---------------------------------------------------------------------------
// WMMA GEMM: C[M,128] = (relu?)(A[M,128] @ W[128,128] + bias[128])
// block = 256 threads (8 waves), each wave = one 16-row tile; 128 rows/block.
// W is held in LDS pair-interleaved so every B fragment is one ds_load_b64
// straight into a WMMA-ready even-aligned VGPR pair. Bias is pulled into LDS
// with the CDNA5 async global->LDS path (ASYNCcnt).
// ---------------------------------------------------------------------------
__global__ __launch_bounds__(256) void gemm128_kernel(
    const float* __restrict__ A, const float* __restrict__ W,
    const float* __restrict__ bias, float* __restrict__ C, int M, int relu) {
  __shared__ float wl[HID * HID];        // 64 KB, pair-interleaved weight
  __shared__ float bl[HID];              // bias (async-copied)

  // async copy of bias: wave 0, each lane moves 16 bytes (32 x 16B = 512B)
  if (threadIdx.x < 32) {
    unsigned loff = (unsigned)(uintptr_t)&bl[threadIdx.x * 4];
    const float* gsrc = bias + threadIdx.x * 4;
    asm volatile("global_load_async_to_lds_b128 %0, %1, off"
                 :: "v"(loff), "v"(gsrc) : "memory");
    asm volatile("s_wait_asynccnt 0x0" ::: "memory");
  }

  // pair-interleave: wl[((k>>1)*128 + n)*2 + (k&1)] = W[k*128 + n]
  for (int idx = threadIdx.x; idx < HID * HID; idx += 256) {
    int kk = idx >> 7;
    int n  = idx & 127;
    wl[((((kk >> 1) * HID) + n) << 1) | (kk & 1)] = W[idx];
  }
  __syncthreads();

  int wave = threadIdx.x >> 5;
  int lane = threadIdx.x & 31;
  int half = lane >> 4;
  int m    = lane & 15;
  int row0 = (blockIdx.x * 8 + wave) * 16;
  if (row0 >= M) return;                 // wave-uniform

  v8f acc[8];
#pragma unroll
  for (int t = 0; t < 8; ++t) {
    float bv = bl[t * 16 + m];
#pragma unroll
    for (int r = 0; r < 8; ++r) acc[t][r] = bv;
  }

  const v2f* wp = (const v2f*)wl;        // pairs (W[2j][n], W[2j+1][n])
  for (int k = 0; k < HID; k += 4) {
    v2f a;
    const float* ap = A + (size_t)(row0 + m) * HID + k + 2 * half;
    a.x = ap[0]; a.y = ap[1];
    const v2f* wrow = wp + ((k >> 1) + half) * HID + m;
#pragma unroll
    for (int t = 0; t < 8; ++t) {
      v2f b = wrow[t * 16];              // one ds_load_b64 per fragment
      acc[t] = wmma4f32(a, b, acc[t]);
    }
  }

#pragma unroll
  for (int t = 0; t < 8; ++t) {
#pragma unroll
    for (int r = 0; r < 8; ++r) {
      int row = row0 + r + 8 * half;
      float v = acc[t][r];
      if (relu) v = fmaxf(v, 0.f);
      C[(size_t)row * HID + t * 16 + m] = v;
    }
  }
}

// ---------------------------------------------------------------------------
// Graph pieces
// ---------------------------------------------------------------------------
__global__ void degacc_kernel(const int* __restrict__ rows, float* __restrict__ deg) {
  int e = blockIdx.x * blockDim.x + threadIdx.x;
  if (e < NEDGES) atomAddF(&deg[rows[e]], 1.0f);
}

__global__ void dinv_kernel(const float* __restrict__ deg, float* __restrict__ dinv) {
  int v = blockIdx.x * blockDim.x + threadIdx.x;
  if (v < NNODES) dinv[v] = rsqrtf(deg[v]);   // deg >= 1 always
}

// agg[cols[e], :] += dinv[r]*dinv[c] * h2[rows[e], :]  (1 thread per edge x 4 floats)
__global__ void edge_kernel(const int* __restrict__ rows, const int* __restrict__ cols,
                            const float* __restrict__ dinv, const float* __restrict__ h2,
                            float* __restrict__ agg) {
  int tid = blockIdx.x * blockDim.x + threadIdx.x;
  if (tid >= NEDGES * 32) return;
  int e = tid >> 5;
  int f = (tid & 31) * 4;
  // prefetch upcoming edge-index cachelines (global_prefetch_b8)
  if (e + 8192 < NEDGES) {
    __builtin_prefetch(&rows[e + 8192], 0, 1);
    __builtin_prefetch(&cols[e + 8192], 0, 1);
  }
  int r = rows[e], c = cols[e];
  float w = dinv[r] * dinv[c];
  const float* src = h2 + (size_t)r * HID + f;
  float* dst = agg + (size_t)c * HID + f;
#pragma unroll
  for (int i = 0; i < 4; ++i) atomAddF(&dst[i], w * src[i]);
}

// h += relu(agg + dinv^2*h2 + conv_b)   (self-loop term added here)
__global__ void combine_kernel(float* __restrict__ h, const float* __restrict__ agg,
                               const float* __restrict__ h2, const float* __restrict__ dinv,
                               const float* __restrict__ cb) {
  int tid = blockIdx.x * blockDim.x + threadIdx.x;
  if (tid >= NNODES * HID) return;
  int v = tid / HID, f = tid % HID;
  float di = dinv[v];
  float a = agg[tid] + di * di * h2[tid] + cb[f];
  h[tid] = h[tid] + fmaxf(a, 0.f);
}

__global__ void pool_kernel(const float* __restrict__ h, const int* __restrict__ batch,
                            float* __restrict__ g) {
  int tid = blockIdx.x * blockDim.x + threadIdx.x;
  if (tid >= NNODES * HID) return;
  int v = tid / HID, f = tid % HID;
  atomAddF(&g[(size_t)batch[v] * HID + f], h[tid]);
}

__global__ void addxg_kernel(float* __restrict__ g, const float* __restrict__ xg) {
  int tid = blockIdx.x * blockDim.x + threadIdx.x;
  if (tid < NGRAPHS * HID) g[tid] += xg[tid];
}

__global__ void bnapply_kernel(const float* __restrict__ A, const float* __restrict__ sums,
                               const float* __restrict__ gamma, const float* __restrict__ bbeta,
                               int M, float* __restrict__ Out) {
  int tid = blockIdx.x * blockDim.x + threadIdx.x;
  if (tid >= M * HID) return;
  int n = tid % HID;
  float invM = 1.f / (float)M;
  float mean = sums[n] * invM;
  float var  = sums[HID + n] * invM - mean * mean;
  Out[tid] = (A[tid] - mean) * rsqrtf(var + BN_EPS) * gamma[n] + bbeta[n];
}

__global__ void cls_kernel(const float* __restrict__ g3, const float* __restrict__ cls_w,
                           const float* __restrict__ cls_b, float* __restrict__ out) {
  int b = blockIdx.x * blockDim.x + threadIdx.x;
  if (b >= NGRAPHS) return;
  const float* gr = g3 + (size_t)b * HID;
  float z0 = cls_b[0], z1 = cls_b[1];
  for (int k = 0; k < HID; ++k) {
    float v = gr[k];
    z0 += v * cls_w[k * 2 + 0];
    z1 += v * cls_w[k * 2 + 1];
  }
  float mx = fmaxf(z0, z1);
  float lse = mx + logf(expf(z0 - mx) + expf(z1 - mx));
  out[b * 2 + 0] = z0 - lse;
  out[b * 2 + 1] = z1 - lse;
}

// ---------------------------------------------------------------------------
// Stage A: 8-hop U/S/D recurrence, one wave per 16-row tile, fp32 WMMA.
// A operand staged C-layout -> row-major through a private global scratch tile
// (same-wave store->load ordering enforced with s_wait_storecnt 0).
// ---------------------------------------------------------------------------
__device__ __forceinline__ void tile_gemm(const float* __restrict__ scratch, int b0,
                                          const float* __restrict__ W,
                                          const float* __restrict__ bias,
                                          v8f* out8, int lane) {
  int half = lane >> 4, m = lane & 15;
#pragma unroll 1
  for (int t = 0; t < 8; ++t) {
    v8f acc;
    float bv = bias[t * 16 + m];
#pragma unroll
    for (int r = 0; r < 8; ++r) acc[r] = bv;
    for (int k = 0; k < HID; k += 4) {
      v2f a, b;
      const float* ap = scratch + (size_t)(b0 + m) * HID + k + 2 * half;
      a.x = ap[0]; a.y = ap[1];
      b.x = W[(k + 2 * half) * HID + t * 16 + m];
      b.y = W[(k + 2 * half + 1) * HID + t * 16 + m];
      acc = wmma4f32(a, b, acc);
    }
    out8[t] = acc;
  }
}

#define STAGE_TILE(EXPR)                                                        \
  {                                                                             \
    _Pragma("unroll") for (int t = 0; t < 8; ++t) {                             \
      _Pragma("unroll") for (int r = 0; r < 8; ++r) {                           \
        scratch[(size_t)(b0 + r + 8 * half) * HID + t * 16 + m] = (EXPR);       \
      }                                                                         \
    }                                                                           \
    __asm__ volatile("s_wait_storecnt 0x0" ::: "memory");                       \
  }

__global__ __launch_bounds__(32) void hopchain_kernel(
    const float* __restrict__ user_state,
    const float* __restrict__ wu0_w, const float* __restrict__ wu0_b,
    const float* __restrict__ ws0_b, const float* __restrict__ wd0_b,
    const float* __restrict__ wu_w, const float* __restrict__ wu_b,
    const float* __restrict__ ws_w, const float* __restrict__ ws_b,
    const float* __restrict__ wd_w, const float* __restrict__ wd_b,
    const float* __restrict__ ralpha, const float* __restrict__ rbeta,
    float* scratch, float* __restrict__ U, float* __restrict__ S,
    float* __restrict__ D) {
  int lane = threadIdx.x & 31;
  int half = lane >> 4, m = lane & 15;
  int b0 = blockIdx.x * 16;
  float alpha = 1.f / (1.f + expf(-ralpha[0]));
  float beta  = 1.f / (1.f + expf(-rbeta[0]));
  float sc1 = 1.f - alpha - beta;

  // u0 for the 8 rows this lane's C-layout slots cover (redundant across lanes, tiny)
  float u0v[8];
#pragma unroll 1
  for (int r = 0; r < 8; ++r) {
    const float* us = user_state + (size_t)(b0 + r + 8 * half) * 200;
    float s = 0.f;
    for (int j = 0; j < 200; ++j) s += us[j];
    u0v[r] = s;
  }

  v8f su[8], ss[8], sd[8];
#pragma unroll
  for (int t = 0; t < 8; ++t) {
    int n = t * 16 + m;
    float w0 = wu0_w[n], ub = wu0_b[n], sb = ws0_b[n], db = wd0_b[n];
#pragma unroll
    for (int r = 0; r < 8; ++r) {
      su[t][r] = u0v[r] * w0 + ub;   // u0 @ wu0_w + wu0_b
      ss[t][r] = sb;                 // 0 @ ws0_w + ws0_b
      sd[t][r] = db;
    }
  }

#pragma unroll 1
  for (int hop = 0; hop < MAXHOP; ++hop) {
    // U_ = ((1-a-b)*U_) @ wu_w + wu_b
    STAGE_TILE(sc1 * su[t][r]);
    tile_gemm(scratch, b0, wu_w, wu_b, su, lane);
    // S_ = (S_ + a*U_) @ ws_w + ws_b   (uses updated U_)
    STAGE_TILE(ss[t][r] + alpha * su[t][r]);
    tile_gemm(scratch, b0, ws_w, ws_b, ss, lane);
    // D_ = (D_ + b*U_) @ wd_w + wd_b
    STAGE_TILE(sd[t][r] + beta * su[t][r]);
    tile_gemm(scratch, b0, wd_w, wd_b, sd, lane);
    // store hop slabs: [B, MAXHOP, HID]
#pragma unroll
    for (int t = 0; t < 8; ++t) {
#pragma unroll
      for (int r = 0; r < 8; ++r) {
        size_t base = ((size_t)(b0 + r + 8 * half) * MAXHOP + hop) * HID + t * 16 + m;
        U[base] = su[t][r];
        S[base] = ss[t][r];
        D[base] = sd[t][r];
      }
    }
  }
}

// xg[b,n] = wx_b[n] + sum_k U[b,idx,k]*wx_w[k,n] + S...*wx_w[128+k,n] + D...*wx_w[256+k,n]
__global__ void xg_kernel(const float* __restrict__ U, const float* __restrict__ S,
                          const float* __restrict__ D, const int* __restrict__ n_hop,
                          const float* __restrict__ wx_w, const float* __restrict__ wx_b,
                          float* __restrict__ xg) {
  int tid = blockIdx.x * blockDim.x + threadIdx.x;
  if (tid >= NGRAPHS * HID) return;
  int b = tid / HID, n = tid % HID;
  int idx = n_hop[b] - 1;
  const float* u = U + ((size_t)b * MAXHOP + idx) * HID;
  const float* s = S + ((size_t)b * MAXHOP + idx) * HID;
  const float* d = D + ((size_t)b * MAXHOP + idx) * HID;
  float acc = wx_b[n];
  for (int k = 0; k < HID; ++k) {
    acc += u[k] * wx_w[k * HID + n];
    acc += s[k] * wx_w[(HID + k) * HID + n];
    acc += d[k] * wx_w[(2 * HID + k) * HID + n];
  }
  xg[tid] = acc;
}

// Uo/So/Do: dot each [b,hop,:] with lu/ls/ld; write straight to d_out
__global__ void uso_kernel(const float* __restrict__ U, const float* __restrict__ S,
                           const float* __restrict__ D,
                           const float* __restrict__ lu_w, const float* __restrict__ lu_b,
                           const float* __restrict__ ls_w, const float* __restrict__ ls_b,
                           const float* __restrict__ ld_w, const float* __restrict__ ld_b,
                           float* __restrict__ out) {
  int tid = blockIdx.x * blockDim.x + threadIdx.x;
  if (tid >= NGRAPHS * MAXHOP) return;
  const float* u = U + (size_t)tid * HID;
  const float* s = S + (size_t)tid * HID;
  const float* d = D + (size_t)tid * HID;
  float au = lu_b[0], as = ls_b[0], ad = ld_b[0];
  for (int k = 0; k < HID; ++k) {
    au += u[k] * lu_w[k];
    as += s[k] * ls_w[k];
    ad += d[k] * ld_w[k];
  }
  out[NGRAPHS * 2 + tid]        = au;  // Uo at 1024
  out[NGRAPHS * 2 + 4096 + tid] = as;  // So at 5120
  out[NGRAPHS * 2 + 8192 + tid] = ad;  // Do at 9216
}

// ---------------------------------------------------------------------------
// Launch
// ---------------------------------------------------------------------------
static inline unsigned gb(long n, int b) { return (unsigned)((n + b - 1) / b); }

extern "C" void kernel_launch(void* const* d_in, const int* in_sizes, int n_in,
                              void* d_out, int out_size, void* d_ws, size_t ws_size,
                              hipStream_t stream) {
  const float* x          = (const float*)d_in[0];
  const float* user_state = (const float*)d_in[1];
  const float* bn_feat_g  = (const float*)d_in[2];
  const float* bn_feat_b  = (const float*)d_in[3];
  const float* w_feat     = (const float*)d_in[4];
  const float* bn_g       = (const float*)d_in[5];
  const float* bn_b       = (const float*)d_in[6];
  const float* conv_w     = (const float*)d_in[7];
  const float* conv_b     = (const float*)d_in[8];
  const float* bn_fc_g    = (const float*)d_in[9];
  const float* bn_fc_b    = (const float*)d_in[10];
  const float* lin_w      = (const float*)d_in[11];
  const float* lin_b      = (const float*)d_in[12];
  const float* bn_hid_g   = (const float*)d_in[13];
  const float* bn_hid_b   = (const float*)d_in[14];
  const float* cls_w      = (const float*)d_in[15];
  const float* cls_b      = (const float*)d_in[16];
  const float* wu0_w      = (const float*)d_in[17];
  const float* wu0_b      = (const float*)d_in[18];
  /* ws0_w d_in[19] multiplies zeros -> unused */
  const float* ws0_b      = (const float*)d_in[20];
  /* wd0_w d_in[21] unused */
  const float* wd0_b      = (const float*)d_in[22];
  const float* wu_w       = (const float*)d_in[23];
  const float* wu_b       = (const float*)d_in[24];
  const float* ws_w       = (const float*)d_in[25];
  const float* ws_b       = (const float*)d_in[26];
  const float* wd_w       = (const float*)d_in[27];
  const float* wd_b       = (const float*)d_in[28];
  const float* wx_w       = (const float*)d_in[29];
  const float* wx_b       = (const float*)d_in[30];
  const float* lu_w       = (const float*)d_in[31];
  const float* lu_b       = (const float*)d_in[32];
  const float* ls_w       = (const float*)d_in[33];
  const float* ls_b       = (const float*)d_in[34];
  const float* ld_w       = (const float*)d_in[35];
  const float* ld_b       = (const float*)d_in[36];
  const float* raw_alpha  = (const float*)d_in[37];
  const float* raw_beta   = (const float*)d_in[38];
  const int*   edge_index = (const int*)d_in[39];
  const int*   batch      = (const int*)d_in[40];
  const int*   n_hop      = (const int*)d_in[41];

  const int* rows = edge_index;
  const int* cols = edge_index + NEDGES;
  float* out = (float*)d_out;

  // workspace layout (floats)
  float* wsf = (float*)d_ws;
  size_t off = 0;
  float* h      = wsf + off; off += (size_t)NNODES * HID;   // 6.4M
  float* h2     = wsf + off; off += (size_t)NNODES * HID;
  float* agg    = wsf + off; off += (size_t)NNODES * HID;
  float* U      = wsf + off; off += (size_t)NGRAPHS * MAXHOP * HID;
  float* S      = wsf + off; off += (size_t)NGRAPHS * MAXHOP * HID;
  float* D      = wsf + off; off += (size_t)NGRAPHS * MAXHOP * HID;
  float* scrA   = wsf + off; off += (size_t)NGRAPHS * HID;
  float* xg     = wsf + off; off += (size_t)NGRAPHS * HID;
  float* colsum = wsf + off; off += 2 * HID;
  float* Weff   = wsf + off; off += HID * HID;
  float* beff   = wsf + off; off += HID;
  float* deg    = wsf + off; off += NNODES;
  float* dinv   = wsf + off; off += NNODES;
  float* g      = wsf + off; off += (size_t)NGRAPHS * HID;
  float* g2     = wsf + off; off += (size_t)NGRAPHS * HID;
  float* g3     = wsf + off; off += (size_t)NGRAPHS * HID;

  // ---- Stage A: hop recurrence (WMMA) + Uo/So/Do + xg ----
  hopchain_kernel<<<NGRAPHS / 16, 32, 0, stream>>>(
      user_state, wu0_w, wu0_b, ws0_b, wd0_b, wu_w, wu_b, ws_w, ws_b,
      wd_w, wd_b, raw_alpha, raw_beta, scrA, U, S, D);
  uso_kernel<<<gb(NGRAPHS * MAXHOP, 256), 256, 0, stream>>>(
      U, S, D, lu_w, lu_b, ls_w, ls_b, ld_w, ld_b, out);
  xg_kernel<<<gb(NGRAPHS * HID, 256), 256, 0, stream>>>(U, S, D, n_hop, wx_w, wx_b, xg);

  // ---- GCN degree norm (self loop contributes 1) ----
  fill_kernel<<<gb(NNODES, 256), 256, 0, stream>>>(deg, 1.0f, NNODES);
  degacc_kernel<<<gb(NEDGES, 256), 256, 0, stream>>>(rows, deg);
  dinv_kernel<<<gb(NNODES, 256), 256, 0, stream>>>(deg, dinv);

  // ---- feature layer: h = relu(bn(x) @ w_feat) with BN folded into weights ----
  zero_kernel<<<1, 256, 0, stream>>>(colsum, 2 * HID);
  colreduce_kernel<<<gb(NNODES, 256), 128, 0, stream>>>(x, NNODES, colsum);
  fold_kernel<<<1, 128, 0, stream>>>(colsum, bn_feat_g, bn_feat_b, w_feat, nullptr,
                                     NNODES, Weff, beff);
  gemm128_kernel<<<gb(NNODES, 128), 256, 0, stream>>>(x, Weff, beff, h, NNODES, 1);

  // ---- 3 GCN conv layers ----
  for (int i = 0; i < 3; ++i) {
    zero_kernel<<<1, 256, 0, stream>>>(colsum, 2 * HID);
    colreduce_kernel<<<gb(NNODES, 256), 128, 0, stream>>>(h, NNODES, colsum);
    fold_kernel<<<1, 128, 0, stream>>>(colsum, bn_g + i * HID, bn_b + i * HID,
                                       conv_w + (size_t)i * HID * HID, nullptr,
                                       NNODES, Weff, beff);
    gemm128_kernel<<<gb(NNODES, 128), 256, 0, stream>>>(h, Weff, beff, h2, NNODES, 0);
    zero_kernel<<<gb((long)NNODES * HID, 256), 256, 0, stream>>>(agg, NNODES * HID);
    edge_kernel<<<gb((long)NEDGES * 32, 256), 256, 0, stream>>>(rows, cols, dinv, h2, agg);
    combine_kernel<<<gb((long)NNODES * HID, 256), 256, 0, stream>>>(h, agg, h2, dinv,
                                                                    conv_b + i * HID);
  }

  // ---- global add pool + xg ----
  zero_kernel<<<gb(NGRAPHS * HID, 256), 256, 0, stream>>>(g, NGRAPHS * HID);
  pool_kernel<<<gb((long)NNODES * HID, 256), 256, 0, stream>>>(h, batch, g);
  addxg_kernel<<<gb(NGRAPHS * HID, 256), 256, 0, stream>>>(g, xg);

  // ---- fc: g2 = relu(bn(g) @ lin_w + lin_b), BN folded ----
  zero_kernel<<<1, 256, 0, stream>>>(colsum, 2 * HID);
  colreduce_kernel<<<gb(NGRAPHS, 256), 128, 0, stream>>>(g, NGRAPHS, colsum);
  fold_kernel<<<1, 128, 0, stream>>>(colsum, bn_fc_g, bn_fc_b, lin_w, lin_b,
                                     NGRAPHS, Weff, beff);
  gemm128_kernel<<<gb(NGRAPHS, 128), 256, 0, stream>>>(g, Weff, beff, g2, NGRAPHS, 1);

  // ---- bn_hid + classifier + log_softmax ----
  zero_kernel<<<1, 256, 0, stream>>>(colsum, 2 * HID);
  colreduce_kernel<<<gb(NGRAPHS, 256), 128, 0, stream>>>(g2, NGRAPHS, colsum);
  bnapply_kernel<<<gb(NGRAPHS * HID, 256), 256, 0, stream>>>(g2, colsum, bn_hid_g,
                                                             bn_hid_b, NGRAPHS, g3);
  cls_kernel<<<gb(NGRAPHS, 256), 256, 0, stream>>>(g3, cls_w, cls_b, out);
}